// HHCBR_53077205844379
// MI455X (gfx1250) — compile-verified
//
#include <hip/hip_runtime.h>
#include <math.h>

// CDNA5 / gfx1250, wave32.
typedef __attribute__((ext_vector_type(2))) float v2f;
typedef __attribute__((ext_vector_type(8))) float v8f;

// ---------------------------------------------------------------------------
// init: f = [featA; featB], accA = featA, accB = featB  (float4 element-parallel)
// ---------------------------------------------------------------------------
__global__ __launch_bounds__(256) void k_init_feats(
    const float* __restrict__ fa, const float* __restrict__ fb,
    float* __restrict__ f, float* __restrict__ accA, float* __restrict__ accB,
    int nA, int nB) {
  long i = (long)blockIdx.x * blockDim.x + threadIdx.x;
  long tot = (long)(nA + nB) * 16;        // rows * 64 floats / 4
  if (i >= tot) return;
  long aCnt = (long)nA * 16;
  float4 v;
  if (i < aCnt) {
    v = ((const float4*)fa)[i];
    ((float4*)accA)[i] = v;
  } else {
    v = ((const float4*)fb)[i - aCnt];
    ((float4*)accB)[i - aCnt] = v;
  }
  ((float4*)f)[i] = v;
}

__global__ __launch_bounds__(256) void k_zero(float* __restrict__ y, long n4) {
  long i = (long)blockIdx.x * blockDim.x + threadIdx.x;
  if (i >= n4) return;
  ((float4*)y)[i] = make_float4(0.f, 0.f, 0.f, 0.f);
}

// ---------------------------------------------------------------------------
// Edge-parallel SpMM: one wave per directed edge. Lane l handles feature
// components [2l, 2l+1]: global_load_b64 gather + 2x global_atomic_add_f32.
// Working sets fit in the 192MB L2, so these atomics/gathers are L2-resident.
// idx layout: rows at idx[0..nE), cols at idx[nE..2nE).
// ---------------------------------------------------------------------------
__global__ __launch_bounds__(256) void k_spmm_atomic(
    const int* __restrict__ idx, const float* __restrict__ val,
    const float* __restrict__ x, float* __restrict__ y, int nE) {
  int gid  = blockIdx.x * blockDim.x + threadIdx.x;
  int e    = gid >> 5;
  int lane = threadIdx.x & 31;
  if (e >= nE) return;
  int r = idx[e];
  int c = idx[nE + e];
  float v = val[e];
  const float2 xv = *(const float2*)(x + (size_t)c * 64 + lane * 2);
  float* yp = y + (size_t)r * 64 + lane * 2;
  atomicAdd(yp,     xv.x * v);
  atomicAdd(yp + 1, xv.y * v);
}

// ---------------------------------------------------------------------------
// f = y * scale;  acc += f / max(||f||_2, 1e-12)   (row-wise, D=64)
// One wave handles 16 rows. Row sum-of-squares via V_WMMA_F32_16X16X4_F32:
//   A(16x4) = squared scaled values, B(4x16) = ones  =>  C[r][*] = sumsq(row r)
// A-matrix f32 16x4 VGPR layout (ISA 7.12.2): lanes 0-15 -> K=0,1 ; lanes
// 16-31 -> K=2,3 (2 VGPRs). C/D layout: VGPR v holds rows v (lanes 0-15) and
// v+8 (lanes 16-31). n and nA are multiples of 16, so no tails and EXEC is
// all-ones in every active wave (WMMA requirement).
// ---------------------------------------------------------------------------
__global__ __launch_bounds__(256) void k_scale_norm_acc(
    const float* __restrict__ y, float* __restrict__ f,
    float* __restrict__ accA, float* __restrict__ accB,
    int n, int nA, float scale) {
  int wid  = (blockIdx.x * blockDim.x + threadIdx.x) >> 5;
  int lane = threadIdx.x & 31;
  int row0 = wid * 16;
  if (row0 >= n) return;

  int rA     = row0 + (lane & 15);      // row this lane feeds into A
  int colSel = (lane >> 4) * 2;         // 0 for lanes 0-15, 2 for lanes 16-31

  v8f c = {};
  v2f bones = {1.0f, 1.0f};
#pragma unroll
  for (int kb = 0; kb < 64; kb += 4) {
    float2 t = *(const float2*)(y + (size_t)rA * 64 + kb + colSel);
    t.x *= scale; t.y *= scale;
    // store scaled f (input of the next SpMM layer)
    *(float2*)(f + (size_t)rA * 64 + kb + colSel) = t;
    v2f a;
    a.x = t.x * t.x;
    a.y = t.y * t.y;
    c = __builtin_amdgcn_wmma_f32_16x16x4_f32(
        /*neg_a=*/false, a, /*neg_b=*/false, bones,
        /*c_mod=*/(short)0, c, /*reuse_a=*/false, /*reuse_b=*/false);
  }

  // Broadcast each row's sumsq from the D-matrix lane layout, then update acc.
  // Re-read y (L0-hot) instead of f to avoid same-wave cross-lane store->load.
#pragma unroll
  for (int r = 0; r < 16; ++r) {
    float s   = __shfl(c[r & 7], (r < 8) ? 0 : 16, 32);
    float inv = 1.0f / fmaxf(sqrtf(s), 1e-12f);
    float k   = scale * inv;            // f*inv == y*scale*inv
    int row   = row0 + r;
    const float2 tv = *(const float2*)(y + (size_t)row * 64 + lane * 2);
    float* ap = ((row < nA) ? (accA + (size_t)row * 64)
                            : (accB + (size_t)(row - nA) * 64)) + lane * 2;
    float2 av = *(float2*)ap;
    av.x += tv.x * k;
    av.y += tv.y * k;
    *(float2*)ap = av;
  }
}

// ---------------------------------------------------------------------------
static void run_propagate(const int* idx, const float* val, int nE,
                          const float* featA, int nA,
                          const float* featB, int nB,
                          float* f, float* y, float* accA, float* accB,
                          hipStream_t stream) {
  int n = nA + nB;
  long n4 = (long)n * 16;
  unsigned gInit = (unsigned)((n4 + 255) / 256);
  unsigned gSpmm = (unsigned)((nE + 7) / 8);        // 8 waves (edges) / block
  unsigned gNorm = (unsigned)((n / 16 + 7) / 8);    // 16 rows / wave, 8 waves / block

  k_init_feats<<<gInit, 256, 0, stream>>>(featA, featB, f, accA, accB, nA, nB);

  // num_layers == 2 in the reference setup; scale = 1/(i+2)
  const float scales[2] = {0.5f, 1.0f / 3.0f};
  for (int i = 0; i < 2; ++i) {
    k_zero<<<gInit, 256, 0, stream>>>(y, n4);
    k_spmm_atomic<<<gSpmm, 256, 0, stream>>>(idx, val, f, y, nE);
    k_scale_norm_acc<<<gNorm, 256, 0, stream>>>(y, f, accA, accB, n, nA, scales[i]);
  }
}

extern "C" void kernel_launch(void* const* d_in, const int* in_sizes, int n_in,
                              void* d_out, int out_size, void* d_ws, size_t ws_size,
                              hipStream_t stream) {
  const float* users_f   = (const float*)d_in[0];
  const float* bundles_f = (const float*)d_in[1];
  const float* items_f   = (const float*)d_in[2];
  const int*   ui_idx    = (const int*)d_in[3];
  const float* ui_val    = (const float*)d_in[4];
  const int*   ub_idx    = (const int*)d_in[5];
  const float* ub_val    = (const float*)d_in[6];
  const int*   ubx_idx   = (const int*)d_in[7];
  const float* ubx_val   = (const float*)d_in[8];
  const int*   agg_idx   = (const int*)d_in[9];
  const float* agg_val   = (const float*)d_in[10];

  const int U = in_sizes[0] / 64;
  const int B = in_sizes[1] / 64;
  const int I = in_sizes[2] / 64;
  const int E_ui  = in_sizes[4];
  const int E_ub  = in_sizes[6];
  const int E_ubx = in_sizes[8];
  const int E_agg = in_sizes[10];

  float* out = (float*)d_out;
  float* ws  = (float*)d_ws;

  const int nMax = U + (I > B ? I : B);      // largest concat row count (U+I)
  float* f    = ws;                           // nMax*64 floats
  float* y    = ws + (size_t)nMax * 64;       // nMax*64 floats
  float* il_i = ws + (size_t)2 * nMax * 64;   // I*64 floats (IL_i accumulator)

  // Output row layout: [IL_u | BL_u | XL_u | IL_b | BL_b | XL_b]
  float* IL_u = out;
  float* BL_u = out + (size_t)U * 64;
  float* XL_u = out + (size_t)2 * U * 64;
  float* IL_b = out + (size_t)3 * U * 64;
  float* BL_b = out + ((size_t)3 * U + B) * 64;
  float* XL_b = out + ((size_t)3 * U + 2 * (size_t)B) * 64;

  // item-level propagation over user-item graph (acc split: users -> out, items -> ws)
  run_propagate(ui_idx, ui_val, E_ui, users_f, U, items_f, I,
                f, y, IL_u, il_i, stream);

  // bundle aggregation from items: IL_b = spmm(agg, IL_i)
  {
    long b4 = (long)B * 16;
    k_zero<<<(unsigned)((b4 + 255) / 256), 256, 0, stream>>>(IL_b, b4);
    k_spmm_atomic<<<(unsigned)((E_agg + 7) / 8), 256, 0, stream>>>(
        agg_idx, agg_val, il_i, IL_b, E_agg);
  }

  // bundle-level propagation over user-bundle graph
  run_propagate(ub_idx, ub_val, E_ub, users_f, U, bundles_f, B,
                f, y, BL_u, BL_b, stream);

  // ingredient-augmented user-bundle propagation
  run_propagate(ubx_idx, ubx_val, E_ubx, users_f, U, bundles_f, B,
                f, y, XL_u, XL_b, stream);
}